// AttentionDecoderRNN_27951647162485
// MI455X (gfx1250) — compile-verified
//
#include <hip/hip_runtime.h>
#include <hip/hip_bf16.h>

// ---------------------------------------------------------------------------
// AttentionDecoderRNN for MI455X (gfx1250), fp32 WMMA path.
//
// Key simplification: softmax over a singleton axis == all-ones, so the
// attention context is just features.sum(axis=1), constant over time.
// Pipeline:
//   K0: ctx = sum_l features[b,l,:]; zero h0, c
//   K1: ctx_gates = ctx @ W_ih[:,E:].T + b_ih + b_hh          (64x4096, K=1024)
//   K2: xg[t]     = embed[captions[:,t]] @ W_ih[:,:E].T       (1984x4096, K=512)
//   K3 (x31):     gates = xg[t] + ctx_gates + h @ W_hh.T; LSTM update
//   K4: logits    = hs @ lin_W.T + lin_b                      (1984x32000, K=1024)
// All GEMMs use v_wmma_f32_16x16x4_f32 (16 waves/block -> 64x64 output tile).
// ---------------------------------------------------------------------------

#define kB 64
#define kT 31
#define kE 512
#define kH 1024
#define kG 4096
#define kV 32000
#define kL 64
#define kMaxLen 32
#define LDSP 68   // LDS pitch in floats: keeps rows 8B/16B aligned, avoids conflicts

typedef float v2f __attribute__((ext_vector_type(2)));
typedef float v8f __attribute__((ext_vector_type(8)));

struct TileSmem {
  const float* arow[64];
  float At[64][LDSP];
  float Bt[64][LDSP];
};

// Core 64x64 output tile GEMM: C += A(64xK) * B(KxN=64), where
// A rows come from sm->arow[r] (gather-friendly) and B[k][n] = W[colBase+n][bofs+k]
// (W row-major with row length ldb). 512 threads = 16 waves = 4x4 tiles of 16x16.
__device__ __forceinline__ void gemm_tile_core(TileSmem* sm, const float* W,
                                               int ldb, int bofs, int K,
                                               int colBase, v8f& acc) {
  const int tid  = threadIdx.x;
  const int lane = tid & 31;
  const int wave = tid >> 5;
  const int tr = (wave >> 2) << 4;   // 0,16,32,48
  const int tc = (wave & 3) << 4;    // 0,16,32,48
  const int hi = lane >> 4;          // 0|1
  const int lo = lane & 15;

  for (int k0 = 0; k0 < K; k0 += 64) {
    // Stage A chunk: 64 rows x 16 float4 (1024 float4), 512 threads x 2
#pragma unroll
    for (int i = 0; i < 2; ++i) {
      int idx = tid + i * 512;
      int r = idx >> 4, q = idx & 15;
      *(float4*)&sm->At[r][q * 4] =
          *(const float4*)(sm->arow[r] + k0 + q * 4);
    }
    // Stage B chunk transposed: read W rows (contiguous in k), write Bt[k][n]
#pragma unroll
    for (int i = 0; i < 2; ++i) {
      int idx = tid + i * 512;
      int n = idx >> 4, q = idx & 15;
      float4 v = *(const float4*)(W + (size_t)(colBase + n) * ldb + bofs + k0 + q * 4);
      sm->Bt[q * 4 + 0][n] = v.x;
      sm->Bt[q * 4 + 1][n] = v.y;
      sm->Bt[q * 4 + 2][n] = v.z;
      sm->Bt[q * 4 + 3][n] = v.w;
    }
    __syncthreads();
#pragma unroll
    for (int kk = 0; kk < 64; kk += 4) {
      int ka = kk + (hi << 1);
      // A frag: lane holds A[m][ka], A[m][ka+1]  (m = lo; lanes16-31 take K+2)
      v2f a = *(const v2f*)&sm->At[tr + lo][ka];
      // B frag mirrors the striping: {B[ka][n], B[ka+1][n]}
      v2f b;
      b.x = sm->Bt[ka + 0][tc + lo];
      b.y = sm->Bt[ka + 1][tc + lo];
      acc = __builtin_amdgcn_wmma_f32_16x16x4_f32(false, a, false, b,
                                                  (short)0, acc, false, false);
    }
    __syncthreads();
  }
}

// Store 16x16 f32 D tile: VGPR r -> rows {r, r+8}, N = lane&15.
__device__ __forceinline__ void store_tile(const v8f& acc, float* out, int ldo,
                                           int rowBase, int colBase,
                                           const float* bias) {
  const int tid  = threadIdx.x;
  const int lane = tid & 31;
  const int wave = tid >> 5;
  const int tr = (wave >> 2) << 4;
  const int tc = (wave & 3) << 4;
  const int hi = lane >> 4;
  const int lo = lane & 15;
  int col = colBase + tc + lo;
  float bv = bias ? bias[col] : 0.0f;
#pragma unroll
  for (int r = 0; r < 8; ++r) {
    int row = rowBase + tr + hi * 8 + r;
    out[(size_t)row * ldo + col] = acc[r] + bv;
  }
}

// K0: ctx[b][j] = sum_l features[b][l][j]; h0 = 0; c = 0.
__global__ __launch_bounds__(256) void k_ctx_init(const float* __restrict__ feat,
                                                  float* __restrict__ ctx,
                                                  float* __restrict__ h0,
                                                  float* __restrict__ c) {
  int idx = blockIdx.x * blockDim.x + threadIdx.x;  // 0 .. 64*1024-1
  int b = idx >> 10, j = idx & 1023;
  const float* p = feat + ((size_t)b * kL) * kH + j;
  float s = 0.0f;
#pragma unroll 8
  for (int l = 0; l < kL; ++l) s += p[(size_t)l * kH];
  ctx[idx] = s;
  h0[idx]  = 0.0f;
  c[idx]   = 0.0f;
}

// K1: ctx_gates = ctx @ W_ih[:, E:].T + (b_ih + b_hh). 64 blocks (N=4096).
__global__ __launch_bounds__(512) void k_ctx_gates(const float* __restrict__ ctx,
                                                   const float* __restrict__ W_ih,
                                                   const float* __restrict__ b_ih,
                                                   const float* __restrict__ b_hh,
                                                   float* __restrict__ cg) {
  __shared__ TileSmem sm;
  int colBase = blockIdx.x * 64;
  if (threadIdx.x < 64) sm.arow[threadIdx.x] = ctx + (size_t)threadIdx.x * kH;
  __syncthreads();
  v8f acc = {};
  gemm_tile_core(&sm, W_ih, kE + kH, kE, kH, colBase, acc);

  const int lane = threadIdx.x & 31, wave = threadIdx.x >> 5;
  const int tr = (wave >> 2) << 4, tc = (wave & 3) << 4;
  const int hi = lane >> 4, lo = lane & 15;
  int col = colBase + tc + lo;
  float bv = b_ih[col] + b_hh[col];
#pragma unroll
  for (int r = 0; r < 8; ++r) {
    int row = tr + hi * 8 + r;
    cg[(size_t)row * kG + col] = acc[r] + bv;
  }
}

// K2: xg[t*B+b] = embed[captions[b][t]] @ W_ih[:, :E].T. Grid (4096/64, 31).
__global__ __launch_bounds__(512) void k_xgates(const float* __restrict__ embed,
                                                const int* __restrict__ captions,
                                                const float* __restrict__ W_ih,
                                                float* __restrict__ xg) {
  __shared__ TileSmem sm;
  int colBase = blockIdx.x * 64;
  int t = blockIdx.y;
  if (threadIdx.x < 64) {
    int cap = captions[threadIdx.x * kMaxLen + t];
    sm.arow[threadIdx.x] = embed + (size_t)cap * kE;
  }
  __syncthreads();
  v8f acc = {};
  gemm_tile_core(&sm, W_ih, kE + kH, 0, kE, colBase, acc);
  store_tile(acc, xg, kG, t * kB, colBase, nullptr);
}

// K3: one LSTM step. 16 blocks; block owns gate columns [jc, jc+64) of each of
// the 4 gates (fused in-register -> nonlinearity needs no cross-block data).
// h is double-buffered across launches; c columns are block-exclusive.
__global__ __launch_bounds__(512) void k_lstm_step(const float* __restrict__ hin,
                                                   float* __restrict__ hout,
                                                   float* __restrict__ cbuf,
                                                   const float* __restrict__ xg,
                                                   const float* __restrict__ cg,
                                                   const float* __restrict__ W_hh,
                                                   float* __restrict__ hs, int t) {
  __shared__ float At[64][LDSP];
  __shared__ float Bt[2][64][LDSP];

  const int jc   = blockIdx.x * 64;         // 0..1023 step 64
  const int tid  = threadIdx.x;
  const int lane = tid & 31;
  const int wave = tid >> 5;
  const int tr = (wave >> 2) << 4;
  const int tc = (wave & 3) << 4;
  const int hi = lane >> 4;
  const int lo = lane & 15;

  v8f acc[4];
  // Two passes over gate pairs to keep static LDS ~52KB.
  for (int gp = 0; gp < 2; ++gp) {
    v8f a0 = {}, a1 = {};
    for (int k0 = 0; k0 < kH; k0 += 64) {
#pragma unroll
      for (int i = 0; i < 2; ++i) {
        int idx = tid + i * 512;
        int r = idx >> 4, q = idx & 15;
        *(float4*)&At[r][q * 4] =
            *(const float4*)(hin + (size_t)r * kH + k0 + q * 4);
      }
#pragma unroll
      for (int i = 0; i < 4; ++i) {
        int idx = tid + i * 512;          // 2 gates x 64 n x 16 q
        int g = idx >> 10;
        int rem = idx & 1023;
        int n = rem >> 4, q = rem & 15;
        int grow = (gp * 2 + g) * kH + jc + n;   // W_hh row index
        float4 v = *(const float4*)(W_hh + (size_t)grow * kH + k0 + q * 4);
        Bt[g][q * 4 + 0][n] = v.x;
        Bt[g][q * 4 + 1][n] = v.y;
        Bt[g][q * 4 + 2][n] = v.z;
        Bt[g][q * 4 + 3][n] = v.w;
      }
      __syncthreads();
#pragma unroll
      for (int kk = 0; kk < 64; kk += 4) {
        int ka = kk + (hi << 1);
        v2f a = *(const v2f*)&At[tr + lo][ka];
        v2f b0; b0.x = Bt[0][ka + 0][tc + lo]; b0.y = Bt[0][ka + 1][tc + lo];
        a0 = __builtin_amdgcn_wmma_f32_16x16x4_f32(false, a, false, b0,
                                                   (short)0, a0, false, false);
        v2f b1; b1.x = Bt[1][ka + 0][tc + lo]; b1.y = Bt[1][ka + 1][tc + lo];
        a1 = __builtin_amdgcn_wmma_f32_16x16x4_f32(false, a, false, b1,
                                                   (short)0, a1, false, false);
      }
      __syncthreads();
    }
    acc[gp * 2 + 0] = a0;
    acc[gp * 2 + 1] = a1;
  }

  // Elementwise LSTM update; each lane owns (b, j) pairs aligned across gates.
  int j = jc + tc + lo;                     // hidden index 0..1023
#pragma unroll
  for (int r = 0; r < 8; ++r) {
    int b = tr + hi * 8 + r;                // batch 0..63
    size_t gi = (size_t)b * kG + j;
    size_t xi = ((size_t)t * kB + b) * kG + j;
    float iv = acc[0][r] + xg[xi + 0 * kH] + cg[gi + 0 * kH];
    float fv = acc[1][r] + xg[xi + 1 * kH] + cg[gi + 1 * kH];
    float gv = acc[2][r] + xg[xi + 2 * kH] + cg[gi + 2 * kH];
    float ov = acc[3][r] + xg[xi + 3 * kH] + cg[gi + 3 * kH];
    float si = 1.0f / (1.0f + __expf(-iv));
    float sf = 1.0f / (1.0f + __expf(-fv));
    float so = 1.0f / (1.0f + __expf(-ov));
    float tg = tanhf(gv);
    size_t ci = (size_t)b * kH + j;
    float cn = sf * cbuf[ci] + si * tg;
    cbuf[ci] = cn;
    float hn = so * tanhf(cn);
    hout[ci] = hn;
    hs[((size_t)t * kB + b) * kH + j] = hn;
  }
}

// K4: logits = hs @ lin_W.T + lin_b. Grid (32000/64, 31) = 15500 blocks.
__global__ __launch_bounds__(512) void k_logits(const float* __restrict__ hs,
                                                const float* __restrict__ lin_W,
                                                const float* __restrict__ lin_b,
                                                float* __restrict__ out) {
  __shared__ TileSmem sm;
  int colBase = blockIdx.x * 64;
  int rowBase = blockIdx.y * 64;
  if (threadIdx.x < 64)
    sm.arow[threadIdx.x] = hs + (size_t)(rowBase + threadIdx.x) * kH;
  __syncthreads();
  v8f acc = {};
  gemm_tile_core(&sm, lin_W, kH, 0, kH, colBase, acc);
  store_tile(acc, out, kV, rowBase, colBase, lin_b);
}

extern "C" void kernel_launch(void* const* d_in, const int* in_sizes, int n_in,
                              void* d_out, int out_size, void* d_ws, size_t ws_size,
                              hipStream_t stream) {
  (void)in_sizes; (void)n_in; (void)out_size; (void)ws_size;
  const float* features = (const float*)d_in[0];
  const int*   captions = (const int*)d_in[1];
  // d_in[2] (lengths) unused: full lengths == time-major flatten.
  const float* embed    = (const float*)d_in[3];
  const float* W_ih     = (const float*)d_in[4];
  const float* W_hh     = (const float*)d_in[5];
  const float* b_ih     = (const float*)d_in[6];
  const float* b_hh     = (const float*)d_in[7];
  // d_in[8], d_in[9] (attn_W/attn_b) unused: singleton softmax -> all-ones.
  const float* lin_W    = (const float*)d_in[10];
  const float* lin_b    = (const float*)d_in[11];
  float* out = (float*)d_out;

  float* ws  = (float*)d_ws;
  float* ctx = ws;  ws += (size_t)kB * kH;       // 64x1024
  float* cg  = ws;  ws += (size_t)kB * kG;       // 64x4096
  float* xg  = ws;  ws += (size_t)kT * kB * kG;  // 1984x4096
  float* h0  = ws;  ws += (size_t)kB * kH;
  float* h1  = ws;  ws += (size_t)kB * kH;
  float* cst = ws;  ws += (size_t)kB * kH;
  float* hs  = ws;  ws += (size_t)kT * kB * kH;  // 1984x1024

  k_ctx_init<<<(kB * kH) / 256, 256, 0, stream>>>(features, ctx, h0, cst);
  k_ctx_gates<<<kG / 64, 512, 0, stream>>>(ctx, W_ih, b_ih, b_hh, cg);
  k_xgates<<<dim3(kG / 64, kT), 512, 0, stream>>>(embed, captions, W_ih, xg);
  for (int t = 0; t < kT; ++t) {
    const float* hin = (t & 1) ? h1 : h0;
    float*       hot = (t & 1) ? h0 : h1;
    k_lstm_step<<<kH / 64, 512, 0, stream>>>(hin, hot, cst, xg, cg, W_hh, hs, t);
  }
  k_logits<<<dim3(kV / 64, kT), 512, 0, stream>>>(hs, lin_W, lin_b, out);
}